// Faster_RCNN_85813446574555
// MI455X (gfx1250) — compile-verified
//
#include <hip/hip_runtime.h>
#include <math.h>

// ---------------------------------------------------------------------------
// Faster-RCNN proposal layer for MI455X (gfx1250, wave32).
//   Kernel 1: decode 147456 anchors -> clipped boxes + positive scores (ws).
//   Kernel 2: persistent 64-WG NMS. Each WG stages its 2304-box slice into
//             LDS with global_load_async_to_lds_b128 (ASYNCcnt), then runs
//             2000 greedy rounds fully LDS-resident with one device barrier
//             per round (double-buffered candidate slots).
// ---------------------------------------------------------------------------

#define FEAT    128
#define NUM_A   9
#define N_TOT   (FEAT * FEAT * NUM_A)   // 147456
#define NWG     64
#define SLICE   (N_TOT / NWG)           // 2304 (multiple of BLK)
#define BLK     256
#define NWAVES  (BLK / 32)
#define MAX_OUT 2000
#define NEGV    (-1e9f)
#define NMS_TH  0.1f

// generate_anchors(16, (1,0.5,2), (1,3,5)) — numpy semantics (banker's round)
__constant__ float c_anchors[NUM_A][4] = {
    {   0.0f,   0.0f,  15.0f,  15.0f },   // r=1.0 s=1
    { -16.0f, -16.0f,  31.0f,  31.0f },   // r=1.0 s=3
    { -32.0f, -32.0f,  47.0f,  47.0f },   // r=1.0 s=5
    {  -3.5f,   2.0f,  18.5f,  13.0f },   // r=0.5 s=1
    { -26.5f, -10.0f,  41.5f,  25.0f },   // r=0.5 s=3
    { -49.5f, -22.0f,  64.5f,  37.0f },   // r=0.5 s=5
    {   2.5f,  -3.0f,  12.5f,  18.0f },   // r=2.0 s=1
    {  -8.5f, -25.0f,  23.5f,  40.0f },   // r=2.0 s=3
    { -19.5f, -47.0f,  34.5f,  62.0f },   // r=2.0 s=5
};

// LDS byte-offset of a __shared__ object (addrspacecast generic->local,
// then ptrtoint gives the in-group offset the async ops want).
typedef __attribute__((address_space(3))) char lds_char_t;
static __device__ __forceinline__ unsigned lds_off(const void* p) {
    return (unsigned)(unsigned long long)(lds_char_t*)p;
}

// CDNA5 async copy: 16 bytes global -> LDS per active lane, tracked on
// ASYNCcnt (no VGPR round trip).
static __device__ __forceinline__ void async_copy_b128(unsigned lds_byte,
                                                       const void* gaddr) {
    asm volatile("global_load_async_to_lds_b128 %0, %1, off"
                 :: "v"(lds_byte), "v"((unsigned long long)gaddr)
                 : "memory");
}

static __device__ __forceinline__ void wait_async0() {
    asm volatile("s_wait_asynccnt 0" ::: "memory");
}

// Monotonic-counter device barrier (persistent kernel, 64 co-resident WGs).
static __device__ __forceinline__ void grid_barrier(int* cnt, int target) {
    __syncthreads();
    if (threadIdx.x == 0) {
        __hip_atomic_fetch_add(cnt, 1, __ATOMIC_ACQ_REL, __HIP_MEMORY_SCOPE_AGENT);
        while (__hip_atomic_load(cnt, __ATOMIC_ACQUIRE, __HIP_MEMORY_SCOPE_AGENT) < target) {
            __builtin_amdgcn_s_sleep(2);
        }
    }
    __syncthreads();
}

// ---------------------------------------------------------------------------
// Kernel 1: anchor decode + clip + score gather.
// ---------------------------------------------------------------------------
__global__ void __launch_bounds__(BLK)
decode_kernel(const float* __restrict__ cls,     // (1,128,128,18)
              const float* __restrict__ bbox,    // (1,128,128,36)
              const int* __restrict__ imh_p,
              const int* __restrict__ imw_p,
              float* __restrict__ boxes,         // (N,4)
              float* __restrict__ scores,        // (N,)
              int* __restrict__ cnt) {
    int i = blockIdx.x * BLK + threadIdx.x;
    if (i == 0) *cnt = 0;                        // re-init barrier each launch
    if (i >= N_TOT) return;

    int a   = i % NUM_A;
    int pos = i / NUM_A;
    int x   = pos & (FEAT - 1);
    int y   = pos >> 7;

    int im_h = imh_p[0], im_w = imw_p[0];
    int sh = im_h >> 7, sw = im_w >> 7;          // stride = im / 128
    float shx = (float)(x * sw + (sw >> 1));
    float shy = (float)(y * sh + (sh >> 1));

    float ax1 = c_anchors[a][0] + shx, ay1 = c_anchors[a][1] + shy;
    float ax2 = c_anchors[a][2] + shx, ay2 = c_anchors[a][3] + shy;
    float w  = ax2 - ax1 + 1.0f, h  = ay2 - ay1 + 1.0f;
    float cx = ax1 + 0.5f * w,   cy = ay1 + 0.5f * h;

    const float* d = bbox + (size_t)i * 4;       // (HW*A,4) layout matches
    float pcx = d[0] * w + cx;
    float pcy = d[1] * h + cy;
    float pw  = expf(d[2]) * w;
    float ph  = expf(d[3]) * h;

    float x1 = pcx - 0.5f * pw, y1 = pcy - 0.5f * ph;
    float x2 = pcx + 0.5f * pw, y2 = pcy + 0.5f * ph;
    float lx = (float)(im_w - 1), ly = (float)(im_h - 1);
    x1 = fminf(fmaxf(x1, 0.0f), lx);
    y1 = fminf(fmaxf(y1, 0.0f), ly);
    x2 = fminf(fmaxf(x2, 0.0f), lx);
    y2 = fminf(fmaxf(y2, 0.0f), ly);

    ((float4*)boxes)[i] = make_float4(x1, y1, x2, y2);
    scores[i] = cls[(size_t)pos * (2 * NUM_A) + NUM_A + a];  // positive half
}

// ---------------------------------------------------------------------------
// Kernel 2: persistent LDS-resident greedy NMS.
// ---------------------------------------------------------------------------
__global__ void __launch_bounds__(BLK)
proposal_nms_kernel(const float* __restrict__ g_boxes,
                    const float* __restrict__ g_scores,
                    float* __restrict__ cand,    // [2][NWG][8] double-buffered
                    int* __restrict__ cnt,
                    float* __restrict__ out) {   // (2000,6)
    __shared__ __align__(16) float4 s_box[SLICE];   // 36 KB
    __shared__ __align__(16) float  s_sc[SLICE];    //  9 KB
    __shared__ float  s_rv[NWAVES];
    __shared__ int    s_ri[NWAVES];
    __shared__ int    s_rg[NWAVES];
    __shared__ float  s_win[6];

    const int wg   = blockIdx.x;
    const int tid  = threadIdx.x;
    const int base = wg * SLICE;

    // ---- Stage this WG's slice into LDS via CDNA5 async path ----
    {
        unsigned bo = lds_off(&s_box[0]);
        const float4* gb = ((const float4*)g_boxes) + base;
        for (int k = tid; k < SLICE; k += BLK)
            async_copy_b128(bo + (unsigned)k * 16u, gb + k);

        unsigned so = lds_off(&s_sc[0]);
        const float4* gs = (const float4*)(g_scores + base);
        for (int k = tid; k < SLICE / 4; k += BLK)
            async_copy_b128(so + (unsigned)k * 16u, gs + k);

        wait_async0();
    }
    __syncthreads();

    const float4* s_sc4 = (const float4*)s_sc;

    int parity = 0;
    for (int it = 0; it < MAX_OUT; ++it) {
        // ---- local argmax over LDS scores, vectorized b128 reads ----
        float bv = -2e9f;
        int   bi = base;
        for (int k4 = tid; k4 < SLICE / 4; k4 += BLK) {
            float4 v = s_sc4[k4];
            int gb4 = base + k4 * 4;
            if (v.x > bv) { bv = v.x; bi = gb4;     }
            if (v.y > bv) { bv = v.y; bi = gb4 + 1; }
            if (v.z > bv) { bv = v.z; bi = gb4 + 2; }
            if (v.w > bv) { bv = v.w; bi = gb4 + 3; }
        }
#pragma unroll
        for (int off = 16; off > 0; off >>= 1) {     // wave32 reduce
            float ov = __shfl_xor(bv, off, 32);
            int   oi = __shfl_xor(bi, off, 32);
            if (ov > bv || (ov == bv && oi < bi)) { bv = ov; bi = oi; }
        }
        if ((tid & 31) == 0) { s_rv[tid >> 5] = bv; s_ri[tid >> 5] = bi; }
        __syncthreads();
        if (tid == 0) {
#pragma unroll
            for (int wv_ = 1; wv_ < NWAVES; ++wv_) {
                float v = s_rv[wv_]; int i2 = s_ri[wv_];
                if (v > bv || (v == bv && i2 < bi)) { bv = v; bi = i2; }
            }
            float4 b = s_box[bi - base];
            float* c = cand + ((size_t)(parity * NWG + wg)) * 8;
            c[0] = bv; c[1] = __int_as_float(bi);
            c[2] = b.x; c[3] = b.y; c[4] = b.z; c[5] = b.w;
        }

        grid_barrier(cnt, (it + 1) * NWG);

        // ---- all-WG candidate reduce: 64 parallel loads + wave reduce ----
        const float* cb = cand + (size_t)parity * NWG * 8;
        {
            float cv = -2e9f; int ci = 0x7fffffff; int cg = 0;
            if (tid < NWG) {
                const float* c = cb + (size_t)tid * 8;
                cv = c[0]; ci = __float_as_int(c[1]); cg = tid;
            }
            if (tid < 64) {                          // waves 0 and 1 only
#pragma unroll
                for (int off = 16; off > 0; off >>= 1) {
                    float ov = __shfl_xor(cv, off, 32);
                    int   oi = __shfl_xor(ci, off, 32);
                    int   og = __shfl_xor(cg, off, 32);
                    if (ov > cv || (ov == cv && oi < ci)) { cv = ov; ci = oi; cg = og; }
                }
                if ((tid & 31) == 0) {
                    s_rv[tid >> 5] = cv; s_ri[tid >> 5] = ci; s_rg[tid >> 5] = cg;
                }
            }
            __syncthreads();
            if (tid == 0) {
                float wv2 = s_rv[0]; int wi2 = s_ri[0]; int g = s_rg[0];
                if (s_rv[1] > wv2 || (s_rv[1] == wv2 && s_ri[1] < wi2)) {
                    wv2 = s_rv[1]; wi2 = s_ri[1]; g = s_rg[1];
                }
                const float* c = cb + (size_t)g * 8;
                s_win[0] = wv2; s_win[1] = __int_as_float(wi2);
                s_win[2] = c[2]; s_win[3] = c[3]; s_win[4] = c[4]; s_win[5] = c[5];
            }
            __syncthreads();
        }

        float wv = s_win[0];
        bool valid = wv > NEGV * 0.5f;
        if (!valid) {
            // scores only decrease: all later rounds invalid too -> fill & exit
            if (wg == 0) {
                for (int r = it + tid; r < MAX_OUT; r += BLK) {
                    float* o = out + (size_t)r * 6;
                    o[0] = 0.0f; o[1] = -1.0f; o[2] = -1.0f;
                    o[3] = -1.0f; o[4] = -1.0f; o[5] = -1.0f;
                }
            }
            break;
        }

        int   wi  = __float_as_int(s_win[1]);
        float wx1 = s_win[2], wy1 = s_win[3], wx2 = s_win[4], wy2 = s_win[5];

        if (wg == 0 && tid == 0) {
            float* o = out + (size_t)it * 6;
            o[0] = 0.0f; o[1] = wx1; o[2] = wy1; o[3] = wx2; o[4] = wy2; o[5] = wv;
        }

        // ---- suppress in LDS: iou > thresh, plus the winner itself ----
        float warea = (wx2 - wx1) * (wy2 - wy1);
        for (int k = tid; k < SLICE; k += BLK) {
            float4 b = s_box[k];
            float ix1 = fmaxf(wx1, b.x), iy1 = fmaxf(wy1, b.y);
            float ix2 = fminf(wx2, b.z), iy2 = fminf(wy2, b.w);
            float inter = fmaxf(ix2 - ix1, 0.0f) * fmaxf(iy2 - iy1, 0.0f);
            float area  = (b.z - b.x) * (b.w - b.y);
            float iou   = inter / (warea + area - inter);
            if (iou > NMS_TH || (base + k) == wi) s_sc[k] = NEGV;
        }
        __syncthreads();
        parity ^= 1;
    }
}

// ---------------------------------------------------------------------------
extern "C" void kernel_launch(void* const* d_in, const int* in_sizes, int n_in,
                              void* d_out, int out_size, void* d_ws, size_t ws_size,
                              hipStream_t stream) {
    (void)in_sizes; (void)n_in; (void)out_size; (void)ws_size;

    const float* cls  = (const float*)d_in[0];   // (1,128,128,18) f32
    const float* bbox = (const float*)d_in[1];   // (1,128,128,36) f32
    const int*   imh  = (const int*)d_in[2];     // scalar
    const int*   imw  = (const int*)d_in[3];     // scalar

    float* boxes  = (float*)d_ws;                         // N*4 f32
    float* scores = boxes + (size_t)N_TOT * 4;            // N   f32
    float* cand   = scores + N_TOT;                       // 2*NWG*8 f32
    int*   cnt    = (int*)(cand + 2 * NWG * 8);           // barrier counter
    float* out    = (float*)d_out;                        // (2000,6) f32

    dim3 dgrid((N_TOT + BLK - 1) / BLK);
    decode_kernel<<<dgrid, BLK, 0, stream>>>(cls, bbox, imh, imw,
                                             boxes, scores, cnt);
    proposal_nms_kernel<<<NWG, BLK, 0, stream>>>(boxes, scores, cand, cnt, out);
}